// ConnectedNormalization_12841952215330
// MI455X (gfx1250) — compile-verified
//
#include <hip/hip_runtime.h>

// ConnectedNormalization, MI455X (gfx1250), wave32.
// Fused single-pass: read inputs once (64 MB), write 69 MB -> ~5.7 us roofline @ 23.3 TB/s.
// k-segment reduction done on the matrix pipe: V_WMMA_F32_16X16X4_F32 with A=ones
// computes column sums of B; B operand built purely by per-lane addressing.

typedef __attribute__((ext_vector_type(2))) float v2f;
typedef __attribute__((ext_vector_type(8))) float v8f;

namespace {
constexpr int N_IN   = 500000;
constexpr int N_NEU  = 20000;
constexpr int BATCH  = 32;
constexpr int KCONN  = 25;           // N_IN / N_NEU
constexpr float EPS  = 1e-6f;
}

__global__ __launch_bounds__(128)
void connorm_fused_wmma(const float* __restrict__ x,     // (N_IN, 32)
                        const float* __restrict__ smean, // (N_NEU,)
                        const float* __restrict__ svar,  // (N_NEU,)
                        float* __restrict__ out)         // (N_IN + 2*N_NEU, 32)
{
    const int wave = (blockIdx.x * (blockDim.x >> 5)) + (threadIdx.x >> 5);
    const int lane = threadIdx.x & 31;
    const int n    = wave;                 // one neuron per wave
    if (n >= N_NEU) return;                // wave-uniform; never taken with exact grid

    const float mu   = smean[n];
    const float rstd = __frsqrt_rn(svar[n] + EPS);
    const float invc = 1.0f / (float)KCONN;

    // A = ones(16x4): every D row == column sums of B.
    v2f aones; aones[0] = 1.0f; aones[1] = 1.0f;

    v8f acc_s0 = {}; v8f acc_q0 = {};      // batch half 0 (b = 0..15): sum, sum^2
    v8f acc_s1 = {}; v8f acc_q1 = {};      // batch half 1 (b = 16..31)

    // B(4x16) f32 layout by pure addressing:
    //   lanes 0-15 : N = lane,    K = 4c+0 (v[0]), 4c+1 (v[1])
    //   lanes 16-31: N = lane-16, K = 4c+2 (v[0]), 4c+3 (v[1])
    // (A=ones makes the sum invariant to the exact K striping.)
    const int nsub = lane & 15;
    const int khi  = (lane >> 4) << 1;     // 0 or 2

    #pragma unroll
    for (int c = 0; c < 6; ++c) {
        const int  k0 = 4 * c + khi;
        const int  k1 = k0 + 1;
        const long r0 = (long)(n + k0 * N_NEU) * BATCH;   // row base, elements
        const long r1 = (long)(n + k1 * N_NEU) * BATCH;

        // half 0: 64B segment per lane-half; half 1: columns +16
        const float v0a = x[r0 + nsub];
        const float v1a = x[r1 + nsub];
        const float v0b = x[r0 + 16 + nsub];
        const float v1b = x[r1 + 16 + nsub];

        v2f b0; b0[0] = v0a;       b0[1] = v1a;
        v2f q0; q0[0] = v0a * v0a; q0[1] = v1a * v1a;
        v2f b1; b1[0] = v0b;       b1[1] = v1b;
        v2f q1; q1[0] = v0b * v0b; q1[1] = v1b * v1b;

        acc_s0 = __builtin_amdgcn_wmma_f32_16x16x4_f32(false, aones, false, b0,
                                                       (short)0, acc_s0, false, false);
        acc_q0 = __builtin_amdgcn_wmma_f32_16x16x4_f32(false, aones, false, q0,
                                                       (short)0, acc_q0, false, false);
        acc_s1 = __builtin_amdgcn_wmma_f32_16x16x4_f32(false, aones, false, b1,
                                                       (short)0, acc_s1, false, false);
        acc_q1 = __builtin_amdgcn_wmma_f32_16x16x4_f32(false, aones, false, q1,
                                                       (short)0, acc_q1, false, false);

        // Emit normalized rows while the data is in registers (stream-once -> NT stores).
        __builtin_nontemporal_store((v0a - mu) * rstd, &out[r0 + nsub]);
        __builtin_nontemporal_store((v1a - mu) * rstd, &out[r1 + nsub]);
        __builtin_nontemporal_store((v0b - mu) * rstd, &out[r0 + 16 + nsub]);
        __builtin_nontemporal_store((v1b - mu) * rstd, &out[r1 + 16 + nsub]);
    }

    // Tail k = 24 on VALU: full 128B coalesced, avoids ragged-K WMMA (0*garbage -> NaN).
    const long rt = (long)(n + (KCONN - 1) * N_NEU) * BATCH;
    const float xt = x[rt + lane];
    __builtin_nontemporal_store((xt - mu) * rstd, &out[rt + lane]);

    // D layout: VGPR0 holds colsum[N] at lane N (M=0) and lane N+16 (M=8) -> no shuffles.
    const float s = (lane < 16 ? acc_s0[0] : acc_s1[0]) + xt;
    const float q = (lane < 16 ? acc_q0[0] : acc_q1[0]) + xt * xt;
    const float m = s * invc;
    const float v = q * invc - m * m;
    out[(long)(N_IN + n) * BATCH + lane]          = m;   // means block
    out[(long)(N_IN + N_NEU + n) * BATCH + lane]  = v;   // variances block
}

extern "C" void kernel_launch(void* const* d_in, const int* in_sizes, int n_in,
                              void* d_out, int out_size, void* d_ws, size_t ws_size,
                              hipStream_t stream) {
    (void)in_sizes; (void)n_in; (void)out_size; (void)d_ws; (void)ws_size;
    const float* x     = (const float*)d_in[0];  // inputs   (N_IN, 32)
    const float* smean = (const float*)d_in[1];  // mean     (N_NEU,)
    const float* svar  = (const float*)d_in[2];  // variance (N_NEU,)
    float* out = (float*)d_out;

    const int threads = 128;                         // 4 waves / block
    const int wavesPerBlock = threads / 32;
    const int blocks = (N_NEU + wavesPerBlock - 1) / wavesPerBlock;  // 5000
    connorm_fused_wmma<<<blocks, threads, 0, stream>>>(x, smean, svar, out);
}